// grad_desc_35923106464232
// MI455X (gfx1250) — compile-verified
//
#include <hip/hip_runtime.h>
#include <hip/hip_bf16.h>

// ---------------------------------------------------------------------------
// SENSE gradient step on MI455X (gfx1250).
//   fftc = L * FFT2{L * x} / 512 with L = (-1)^(h+w)  (exact for N=512, ortho)
//   Line search collapses algebraically to: accept iff 0.5*a*c2 < 0.1*gxNorm,
//   c2 = ||A(gx)||^2 -- so only A(x), At(r), A(gx) are ever computed.
// FFT-512 per wave32: 512 = 16 x 32 Cooley-Tukey, DFT16/DFT32 stages as
// complex matmuls on V_WMMA_F32_16X16X4_F32 (96 WMMA per line).
// All WMMA operand streams are stored so each lane's K-values are CONTIGUOUS
// in LDS (perm1/perm2 groupings + odd row strides 17/33): operands load as
// packed ds_load_b128 with immediate offsets (no repacking movs, few address
// adds, bank-conflict-free).  The 512-point twiddle is applied at stage-2
// operand-load time from a table in the identical layout.  fft512 is split
// into core (accumulators) + store so P2/P5 can fuse elementwise work on
// registers.
// ---------------------------------------------------------------------------

#define NB 4
#define NC 8
#define NHW 512
#define PI_F 3.14159265358979323846f
#define LSTR 544   // padded line-plane size

typedef float v2f __attribute__((ext_vector_type(2)));
typedef float v8f __attribute__((ext_vector_type(8)));

// group K-values per lane-half contiguously: stage-1 (n1, 16 values)
static __device__ __forceinline__ int perm1f(int j) {
    return ((j & 2) >> 1) * 8 + (j >> 2) * 2 + (j & 1);
}
// stage-2 (n2, 32 values)
static __device__ __forceinline__ int perm2f(int n2) {
    return ((n2 & 2) >> 1) * 16 + (n2 >> 2) * 2 + (n2 & 1);
}
// permuted line index: element n -> row (n&31), permuted column
static __device__ __forceinline__ int pidx2(int n) {
    return (n & 31) * 17 + perm1f(n >> 5);
}

static __device__ __forceinline__ v8f v8zero() {
    v8f z;
#pragma unroll
    for (int i = 0; i < 8; ++i) z[i] = 0.f;
    return z;
}

static __device__ __forceinline__ v8f wmma4(v2f a, v2f b, v8f c) {
    // D = A(16x4) * B(4x16) + C(16x16), f32. (neg_a, A, neg_b, B, c_mod, C,
    // reuse_a, reuse_b); F32 NEG only allows CNeg -> negate via VALU xor.
    return __builtin_amdgcn_wmma_f32_16x16x4_f32(false, a, false, b, (short)0, c,
                                                 false, false);
}

// (Cr,Ci) += (Ar + i*Ai)(Br + i*Bi)
static __device__ __forceinline__ void cwmma(v2f ar, v2f ai, v2f br, v2f bi,
                                             v8f& cr, v8f& ci) {
    cr = wmma4(ar, br, cr);
    v2f ain = -ai;
    cr = wmma4(ain, bi, cr);
    ci = wmma4(ar, bi, ci);
    ci = wmma4(ai, br, ci);
}

// Cooperative per-block DFT tables (forward sign; inverse = conjugate on use).
static __device__ void init_tables(float* w16r, float* w16i, float* twr,
                                   float* twi, float* w32r, float* w32i) {
    const int tid = threadIdx.x;
    for (int i = tid; i < 256; i += 256) {           // W16[k1][perm1(n1)]
        int k1 = i >> 4, n1 = i & 15;
        float th = -2.f * PI_F * (float)((k1 * n1) & 15) / 16.f;
        w16r[k1 * 17 + perm1f(n1)] = __cosf(th);
        w16i[k1 * 17 + perm1f(n1)] = __sinf(th);
    }
    for (int i = tid; i < 512; i += 256) {           // TW[k1][perm2(n2)]
        int k1 = i >> 5, n2 = i & 31;                // = w512^(k1*n2)
        float th = -2.f * PI_F * (float)((k1 * n2) & 511) / 512.f;
        twr[k1 * 33 + perm2f(n2)] = __cosf(th);
        twi[k1 * 33 + perm2f(n2)] = __sinf(th);
    }
    for (int i = tid; i < 1024; i += 256) {          // W32T[k2][perm2(n2)]
        int k2 = i >> 5, n2 = i & 31;                // (W32 is symmetric)
        float th = -2.f * PI_F * (float)((n2 * k2) & 31) / 32.f;
        w32r[k2 * 33 + perm2f(n2)] = __cosf(th);
        w32i[k2 * 33 + perm2f(n2)] = __sinf(th);
    }
}

// 512-point FFT core: consumes wave-private permuted SoA line (dr/di), leaves
// result in D-layout accumulators Xr/Xi (element (t2,r) on lane (lm,hb) is
// X[k], k = (r+8*hb) + 16*(16*t2+lm)).  simag: +1 fwd, -1 inverse (unnorm.).
// n = 32*n1 + n2 ; X[k1 + 16*k2] = sum_n2 w512^(n2 k1) w32^(n2 k2)
//                                   * sum_n1 x[32 n1 + n2] w16^(n1 k1)
static __device__ __forceinline__ void fft512_core(
    float* __restrict__ dr, float* __restrict__ di,
    const float* __restrict__ tW16r, const float* __restrict__ tW16i,
    const float* __restrict__ tTWr, const float* __restrict__ tTWi,
    const float* __restrict__ tW32r, const float* __restrict__ tW32i,
    float simag, v8f (&Xr)[2], v8f (&Xi)[2]) {
    const int lane = threadIdx.x & 31;
    const int lm = lane & 15;   // index within lane half
    const int hb = lane >> 4;   // lane half (0/1)

    // ---- stage 1 A: W16, 8 contiguous floats per lane -> 2x b128/plane ----
    v2f Ar16[4], Ai16[4];
    {
        int wbase = lm * 17 + 8 * hb;
#pragma unroll
        for (int c = 0; c < 4; ++c)
#pragma unroll
            for (int v = 0; v < 2; ++v) {
                Ar16[c][v] = tW16r[wbase + 2 * c + v];
                Ai16[c][v] = simag * tW16i[wbase + 2 * c + v];
            }
    }
    // ---- stage 1 matmul: Y[k1, n2]; B contiguous 8 per lane per tile ----
    v8f Yr[2], Yi[2];
#pragma unroll
    for (int t = 0; t < 2; ++t) {                    // n2 half-tiles
        v8f yr = v8zero(), yi = v8zero();
        int base = (16 * t + lm) * 17 + 8 * hb;
#pragma unroll
        for (int c = 0; c < 4; ++c) {
            v2f br, bi;
#pragma unroll
            for (int v = 0; v < 2; ++v) {
                br[v] = dr[base + 2 * c + v];
                bi[v] = di[base + 2 * c + v];
            }
            cwmma(Ar16[c], Ai16[c], br, bi, yr, yi);
        }
        Yr[t] = yr;
        Yi[t] = yi;
    }
    // ---- store raw Y at [k1*33 + perm2(n2)] (scatter; reads were b128) ----
    {
        int p2l = ((lm & 2) >> 1) * 16 + (lm >> 2) * 2 + (lm & 1);  // perm2(lm)
#pragma unroll
        for (int t = 0; t < 2; ++t)
#pragma unroll
            for (int r = 0; r < 8; ++r) {
                int p = (r + 8 * hb) * 33 + p2l + 8 * t;
                dr[p] = Yr[t][r];
                di[p] = Yi[t][r];
            }
    }
    // ---- stage 2 A: 16 contiguous floats per lane (4x b128/plane), with
    //      twiddle w512^(k1*n2) from a table in the SAME layout ----
    v2f Ar2[8], Ai2[8];
    {
        int abase = lm * 33 + 16 * hb;
#pragma unroll
        for (int c2 = 0; c2 < 8; ++c2)
#pragma unroll
            for (int v = 0; v < 2; ++v) {
                float ar = dr[abase + 2 * c2 + v];
                float ai = di[abase + 2 * c2 + v];
                float wr = tTWr[abase + 2 * c2 + v];
                float wi = simag * tTWi[abase + 2 * c2 + v];
                Ar2[c2][v] = ar * wr - ai * wi;
                Ai2[c2][v] = ar * wi + ai * wr;
            }
    }
    // ---- stage 2 matmul: X[k1,k2] = sum_n2 A'[k1,n2] * W32[n2,k2] ----
#pragma unroll
    for (int t2 = 0; t2 < 2; ++t2) {                 // k2 half-tiles
        v8f xr = v8zero(), xi = v8zero();
        int bbase = (16 * t2 + lm) * 33 + 16 * hb;
#pragma unroll
        for (int c2 = 0; c2 < 8; ++c2) {
            v2f br, bi;
#pragma unroll
            for (int v = 0; v < 2; ++v) {
                br[v] = tW32r[bbase + 2 * c2 + v];
                bi[v] = simag * tW32i[bbase + 2 * c2 + v];
            }
            cwmma(Ar2[c2], Ai2[c2], br, bi, xr, xi);
        }
        Xr[t2] = xr;
        Xi[t2] = xi;
    }
}

// Store D-layout accumulators into the permuted line layout (X[k]->pidx2(k)).
static __device__ __forceinline__ void fft512_store(float* __restrict__ dr,
                                                    float* __restrict__ di,
                                                    const v8f (&Xr)[2],
                                                    const v8f (&Xi)[2]) {
    const int lane = threadIdx.x & 31;
    const int lm = lane & 15, hb = lane >> 4;
    const int base0 = (8 * hb + 16 * (lm & 1)) * 17 + perm1f(lm >> 1);
#pragma unroll
    for (int t2 = 0; t2 < 2; ++t2)
#pragma unroll
        for (int r = 0; r < 8; ++r) {
            int p = base0 + r * 17 + 4 * t2;
            dr[p] = Xr[t2][r];
            di[p] = Xi[t2][r];
        }
}

static __device__ __forceinline__ void fft512_wave(
    float* dr, float* di, const float* tW16r, const float* tW16i,
    const float* tTWr, const float* tTWi, const float* tW32r,
    const float* tW32i, float simag) {
    v8f Xr[2], Xi[2];
    fft512_core(dr, di, tW16r, tW16i, tTWr, tTWi, tW32r, tW32i, simag, Xr, Xi);
    fft512_store(dr, di, Xr, Xi);
}

// ---------------------------------------------------------------------------
// P1: y = L(h+w) * x * s  -> row FFT (w) -> W1[b][c][h][kw] (coalesced rows)
// ---------------------------------------------------------------------------
__global__ __launch_bounds__(256) void k_p1(const float* __restrict__ xr,
                                            const float* __restrict__ xi,
                                            const float* __restrict__ sr,
                                            const float* __restrict__ si,
                                            float2* __restrict__ W1) {
    __shared__ float tW16r[272], tW16i[272], tTWr[544], tTWi[544];
    __shared__ float tW32r[1056], tW32i[1056];
    __shared__ float2 xline[512];
    __shared__ float cbr[8][LSTR], cbi[8][LSTR];
    init_tables(tW16r, tW16i, tTWr, tTWi, tW32r, tW32i);
    const int b = blockIdx.x >> 9, h = blockIdx.x & 511;
    for (int w = threadIdx.x; w < 512; w += 256) {
        size_t idx = ((size_t)(b * 512 + h)) * 512 + w;
        xline[w] = make_float2(xr[idx], xi[idx]);
    }
    __syncthreads();
    const int c = threadIdx.x >> 5, lane = threadIdx.x & 31;
    float* myr = cbr[c];
    float* myi = cbi[c];
#pragma unroll 4
    for (int j = 0; j < 16; ++j) {
        int w = lane + 32 * j;
        int p = lane * 17 + perm1f(j);               // pidx2(w)
        float sgn = ((h + w) & 1) ? -1.f : 1.f;
        size_t sidx = (((size_t)(b * 512 + h) * 512 + w) * 8 + c);
        float a = sr[sidx], bb = si[sidx];
        float2 xv = xline[w];
        myr[p] = sgn * (xv.x * a - xv.y * bb);
        myi[p] = sgn * (xv.x * bb + xv.y * a);
    }
    fft512_wave(myr, myi, tW16r, tW16i, tTWr, tTWi, tW32r, tW32i, 1.f);
    float2* out = W1 + (((size_t)(b * 8 + c)) << 18);
#pragma unroll 4
    for (int j = 0; j < 16; ++j) {
        int kw = lane + 32 * j;
        int p = lane * 17 + perm1f(j);
        out[(size_t)h * 512 + kw] = make_float2(myr[p], myi[p]);
    }
}

// ---------------------------------------------------------------------------
// P2: per (b,c, 8 columns): col FFT (h) -> u = mask*F/512 - L*mask*b applied
//     ON THE ACCUMULATOR REGISTERS -> store permuted -> inverse col FFT ->
//     in-place tile store (block owns its 8 columns; no races).
// ---------------------------------------------------------------------------
__global__ __launch_bounds__(256) void k_p2(const float* __restrict__ mk,
                                            const float* __restrict__ br_,
                                            const float* __restrict__ bi_,
                                            float2* __restrict__ W1) {
    __shared__ float tW16r[272], tW16i[272], tTWr[544], tTWi[544];
    __shared__ float tW32r[1056], tW32i[1056];
    __shared__ float cbr[8][LSTR], cbi[8][LSTR];
    init_tables(tW16r, tW16i, tTWr, tTWi, tW32r, tW32i);
    const int b = blockIdx.x >> 9, c = (blockIdx.x >> 6) & 7, kwg = blockIdx.x & 63;
    float2* img = W1 + (((size_t)(b * 8 + c)) << 18);
    const int kw0 = kwg * 8;
    for (int i = threadIdx.x; i < 4096; i += 256) {  // 64B-chunk column gather
        int u = i >> 3, voff = i & 7;
        float2 d = img[(size_t)u * 512 + kw0 + voff];
        int p = pidx2(u);
        cbr[voff][p] = d.x;
        cbi[voff][p] = d.y;
    }
    __syncthreads();
    const int wv = threadIdx.x >> 5, lane = threadIdx.x & 31;
    const int lm = lane & 15, hb = lane >> 4;
    float* myr = cbr[wv];
    float* myi = cbi[wv];
    const int kw = kw0 + wv;
    // Prefetch this lane's stride-16KB b-gather (one element per cacheline)
    // so the forward FFT hides its latency; lowers to global_prefetch_b8.
#pragma unroll
    for (int t2 = 0; t2 < 2; ++t2)
#pragma unroll
        for (int r = 0; r < 8; ++r) {
            int kh = r + 8 * hb + 16 * lm + 256 * t2;
            size_t bidx = (((size_t)(b * 512 + kh) * 512 + kw) * 8 + c);
            __builtin_prefetch(br_ + bidx, 0, 1);
            __builtin_prefetch(bi_ + bidx, 0, 1);
        }
    v8f Xr[2], Xi[2];
    fft512_core(myr, myi, tW16r, tW16i, tTWr, tTWi, tW32r, tW32i, 1.f, Xr, Xi);
    const float inv512 = 1.f / 512.f;
#pragma unroll
    for (int t2 = 0; t2 < 2; ++t2)
#pragma unroll
        for (int r = 0; r < 8; ++r) {
            int kh = r + 8 * hb + 16 * lm + 256 * t2;    // this element's row
            float m = mk[kh * 512 + kw];
            float sgn = ((kh + kw) & 1) ? -1.f : 1.f;
            size_t bidx = (((size_t)(b * 512 + kh) * 512 + kw) * 8 + c);
            Xr[t2][r] = m * (Xr[t2][r] * inv512 - sgn * br_[bidx]);
            Xi[t2][r] = m * (Xi[t2][r] * inv512 - sgn * bi_[bidx]);
        }
    fft512_store(myr, myi, Xr, Xi);
    fft512_wave(myr, myi, tW16r, tW16i, tTWr, tTWi, tW32r, tW32i, -1.f);
    __syncthreads();
    for (int i = threadIdx.x; i < 4096; i += 256) {
        int u = i >> 3, voff = i & 7;
        int p = pidx2(u);
        img[(size_t)u * 512 + kw0 + voff] = make_float2(cbr[voff][p], cbi[voff][p]);
    }
}

// ---------------------------------------------------------------------------
// P3: inverse row FFT, g = L*(1/512)*conj(s)*(.), sum over channels -> gx,
//     deterministic per-block partial of ||gx||^2.
// ---------------------------------------------------------------------------
__global__ __launch_bounds__(256) void k_p3(const float* __restrict__ sr,
                                            const float* __restrict__ si,
                                            const float2* __restrict__ W1,
                                            float2* __restrict__ W2,
                                            float* __restrict__ R1) {
    __shared__ float tW16r[272], tW16i[272], tTWr[544], tTWi[544];
    __shared__ float tW32r[1056], tW32i[1056];
    __shared__ float cbr[8][LSTR], cbi[8][LSTR];
    __shared__ float red[256];
    init_tables(tW16r, tW16i, tTWr, tTWi, tW32r, tW32i);
    const int b = blockIdx.x >> 9, hp = blockIdx.x & 511;
    const int wv = threadIdx.x >> 5, lane = threadIdx.x & 31;
    const float2* img = W1 + (((size_t)(b * 8 + wv)) << 18);
    float* myr = cbr[wv];
    float* myi = cbi[wv];
#pragma unroll 4
    for (int j = 0; j < 16; ++j) {
        int kw = lane + 32 * j;
        int p = lane * 17 + perm1f(j);
        float2 d = img[(size_t)hp * 512 + kw];
        myr[p] = d.x;
        myi[p] = d.y;
    }
    __syncthreads();
    fft512_wave(myr, myi, tW16r, tW16i, tTWr, tTWi, tW32r, tW32i, -1.f);
    const float inv512 = 1.f / 512.f;
#pragma unroll 4
    for (int j = 0; j < 16; ++j) {
        int wp = lane + 32 * j;
        int p = lane * 17 + perm1f(j);
        float sgn = ((hp + wp) & 1) ? -1.f : 1.f;
        float gx = sgn * inv512 * myr[p], gy = sgn * inv512 * myi[p];
        size_t sidx = (((size_t)(b * 512 + hp) * 512 + wp) * 8 + wv);
        float a = sr[sidx], bb = si[sidx];
        myr[p] = a * gx + bb * gy;                   // conj(s) * g
        myi[p] = a * gy - bb * gx;
    }
    __syncthreads();
    float part = 0.f;
    for (int w = threadIdx.x; w < 512; w += 256) {
        int p = pidx2(w);
        float rx = 0.f, ry = 0.f;
#pragma unroll
        for (int c = 0; c < 8; ++c) {
            rx += cbr[c][p];
            ry += cbi[c][p];
        }
        W2[((size_t)(b * 512 + hp)) * 512 + w] = make_float2(rx, ry);
        part += rx * rx + ry * ry;
    }
    red[threadIdx.x] = part;
    __syncthreads();
    for (int s = 128; s > 0; s >>= 1) {
        if (threadIdx.x < (unsigned)s) red[threadIdx.x] += red[threadIdx.x + s];
        __syncthreads();
    }
    if (threadIdx.x == 0) R1[blockIdx.x] = red[0];
}

// ---------------------------------------------------------------------------
// P4: same as P1 with gx as the image.
// ---------------------------------------------------------------------------
__global__ __launch_bounds__(256) void k_p4(const float* __restrict__ sr,
                                            const float* __restrict__ si,
                                            const float2* __restrict__ W2,
                                            float2* __restrict__ W1) {
    __shared__ float tW16r[272], tW16i[272], tTWr[544], tTWi[544];
    __shared__ float tW32r[1056], tW32i[1056];
    __shared__ float2 gline[512];
    __shared__ float cbr[8][LSTR], cbi[8][LSTR];
    init_tables(tW16r, tW16i, tTWr, tTWi, tW32r, tW32i);
    const int b = blockIdx.x >> 9, h = blockIdx.x & 511;
    for (int w = threadIdx.x; w < 512; w += 256)
        gline[w] = W2[((size_t)(b * 512 + h)) * 512 + w];
    __syncthreads();
    const int c = threadIdx.x >> 5, lane = threadIdx.x & 31;
    float* myr = cbr[c];
    float* myi = cbi[c];
#pragma unroll 4
    for (int j = 0; j < 16; ++j) {
        int w = lane + 32 * j;
        int p = lane * 17 + perm1f(j);
        float sgn = ((h + w) & 1) ? -1.f : 1.f;
        size_t sidx = (((size_t)(b * 512 + h) * 512 + w) * 8 + c);
        float a = sr[sidx], bb = si[sidx];
        float2 g = gline[w];
        myr[p] = sgn * (g.x * a - g.y * bb);
        myi[p] = sgn * (g.x * bb + g.y * a);
    }
    fft512_wave(myr, myi, tW16r, tW16i, tTWr, tTWi, tW32r, tW32i, 1.f);
    float2* out = W1 + (((size_t)(b * 8 + c)) << 18);
#pragma unroll 4
    for (int j = 0; j < 16; ++j) {
        int kw = lane + 32 * j;
        int p = lane * 17 + perm1f(j);
        out[(size_t)h * 512 + kw] = make_float2(myr[p], myi[p]);
    }
}

// ---------------------------------------------------------------------------
// P5: col FFT, c2 partial = sum mask * |F|^2 / 512^2 taken DIRECTLY from the
//     accumulator registers (no store).
// ---------------------------------------------------------------------------
__global__ __launch_bounds__(256) void k_p5(const float* __restrict__ mk,
                                            const float2* __restrict__ W1,
                                            float* __restrict__ R2) {
    __shared__ float tW16r[272], tW16i[272], tTWr[544], tTWi[544];
    __shared__ float tW32r[1056], tW32i[1056];
    __shared__ float cbr[8][LSTR], cbi[8][LSTR];
    __shared__ float red[256];
    init_tables(tW16r, tW16i, tTWr, tTWi, tW32r, tW32i);
    const int b = blockIdx.x >> 9, c = (blockIdx.x >> 6) & 7, kwg = blockIdx.x & 63;
    const float2* img = W1 + (((size_t)(b * 8 + c)) << 18);
    const int kw0 = kwg * 8;
    for (int i = threadIdx.x; i < 4096; i += 256) {
        int u = i >> 3, voff = i & 7;
        float2 d = img[(size_t)u * 512 + kw0 + voff];
        int p = pidx2(u);
        cbr[voff][p] = d.x;
        cbi[voff][p] = d.y;
    }
    __syncthreads();
    const int wv = threadIdx.x >> 5, lane = threadIdx.x & 31;
    const int lm = lane & 15, hb = lane >> 4;
    v8f Xr[2], Xi[2];
    fft512_core(cbr[wv], cbi[wv], tW16r, tW16i, tTWr, tTWi, tW32r, tW32i, 1.f,
                Xr, Xi);
    const int kw = kw0 + wv;
    const float inv2 = (1.f / 512.f) * (1.f / 512.f);
    float part = 0.f;
#pragma unroll
    for (int t2 = 0; t2 < 2; ++t2)
#pragma unroll
        for (int r = 0; r < 8; ++r) {
            int kh = r + 8 * hb + 16 * lm + 256 * t2;
            float fr = Xr[t2][r], fi = Xi[t2][r];
            part += mk[kh * 512 + kw] * (fr * fr + fi * fi);
        }
    red[threadIdx.x] = part * inv2;
    __syncthreads();
    for (int s = 128; s > 0; s >>= 1) {
        if (threadIdx.x < (unsigned)s) red[threadIdx.x] += red[threadIdx.x + s];
        __syncthreads();
    }
    if (threadIdx.x == 0) R2[blockIdx.x] = red[0];
}

// ---------------------------------------------------------------------------
// P6: finalize reductions + scalar Armijo loop (exact reference semantics).
// ---------------------------------------------------------------------------
__global__ void k_p6(const float* __restrict__ R1, const float* __restrict__ R2,
                     float* __restrict__ alpha_out) {
    if (threadIdx.x != 0 || blockIdx.x != 0) return;
    float gn = 0.f, c2 = 0.f;
    for (int i = 0; i < 2048; ++i) gn += R1[i];
    for (int i = 0; i < 2048; ++i) c2 += R2[i];
    float alpha = 0.5f, alpha_used = 0.5f;
    for (int it = 0; it < 250; ++it) {
        // obj_new < obj_x - a*0.9*gn   <=>   0.5*a*c2 < 0.1*gn
        bool accept = (0.5f * alpha * c2) < (0.1f * gn);
        alpha_used = alpha;
        if (accept) break;
        alpha *= 0.9f;
    }
    alpha_out[0] = alpha_used;
}

// ---------------------------------------------------------------------------
// P7: out = [Re(x - a*gx); Im(x - a*gx)]   shape [2,B,H,W]
// ---------------------------------------------------------------------------
__global__ __launch_bounds__(256) void k_p7(const float* __restrict__ xr,
                                            const float* __restrict__ xi,
                                            const float2* __restrict__ W2,
                                            const float* __restrict__ alpha_in,
                                            float* __restrict__ out) {
    int i = blockIdx.x * 256 + threadIdx.x;
    if (i >= (1 << 20)) return;
    float a = alpha_in[0];
    float2 g = W2[i];
    out[i] = xr[i] - a * g.x;
    out[(1 << 20) + i] = xi[i] - a * g.y;
}

// ---------------------------------------------------------------------------
extern "C" void kernel_launch(void* const* d_in, const int* in_sizes, int n_in,
                              void* d_out, int out_size, void* d_ws, size_t ws_size,
                              hipStream_t stream) {
    (void)in_sizes; (void)n_in; (void)out_size; (void)ws_size;
    const float* xr = (const float*)d_in[0];
    const float* xi = (const float*)d_in[1];
    const float* sr = (const float*)d_in[2];
    const float* si = (const float*)d_in[3];
    const float* mk = (const float*)d_in[4];
    const float* br = (const float*)d_in[5];
    const float* bi = (const float*)d_in[6];
    float* out = (float*)d_out;

    char* ws = (char*)d_ws;
    float* alpha = (float*)ws;                       // 1 float
    float* R1 = (float*)(ws + 1024);                 // 2048 floats (gxNorm parts)
    float* R2 = (float*)(ws + 9216);                 // 2048 floats (c2 parts)
    float2* W1 = (float2*)(ws + 32768);              // [B][C][512][512] = 64 MB
    float2* W2 = (float2*)(ws + 32768 + (size_t)NB * NC * NHW * NHW * sizeof(float2));
    // total ws use: ~75.6 MB

    k_p1<<<dim3(NB * NHW), dim3(256), 0, stream>>>(xr, xi, sr, si, W1);
    k_p2<<<dim3(NB * NC * (NHW / 8)), dim3(256), 0, stream>>>(mk, br, bi, W1);
    k_p3<<<dim3(NB * NHW), dim3(256), 0, stream>>>(sr, si, W1, W2, R1);
    k_p4<<<dim3(NB * NHW), dim3(256), 0, stream>>>(sr, si, W2, W1);
    k_p5<<<dim3(NB * NC * (NHW / 8)), dim3(256), 0, stream>>>(mk, W1, R2);
    k_p6<<<dim3(1), dim3(32), 0, stream>>>(R1, R2, alpha);
    k_p7<<<dim3((1 << 20) / 256), dim3(256), 0, stream>>>(xr, xi, W2, alpha, out);
}